// SpanFSED_11029476016243
// MI455X (gfx1250) — compile-verified
//
#include <hip/hip_runtime.h>
#include <math.h>

#define B_ 32
#define S_ 512
#define H_ 768
#define D_ 64
#define N2D_ 128
#define BIGC 1000000000000.0f

typedef float v2f __attribute__((ext_vector_type(2)));
typedef float v8f __attribute__((ext_vector_type(8)));

// ---------------------------------------------------------------------------
// Kernel 1: projection GEMM (M=16384,K=768,N=128) + bias + RoPE.
// One wave (32 threads) computes a 16x128 output tile via 8 WMMA chains.
// Outputs: qw row-major [B*S, 64]; kw transposed kwT [B, 64, 512].
// ---------------------------------------------------------------------------
__global__ __launch_bounds__(32) void proj_rope_kernel(
    const float* __restrict__ hidden,  // [B*S, H]
    const float* __restrict__ W,       // [H, 128]
    const float* __restrict__ bias,    // [128]
    float* __restrict__ qw,            // [B*S, 64]
    float* __restrict__ kwT)           // [B, 64, 512]
{
    const int lane = threadIdx.x;
    const int ln   = lane & 15;
    const int hi   = lane >> 4;          // 0: K=0,1 / M=0..7 ; 1: K=2,3 / M=8..15
    const int m0   = blockIdx.x * 16;    // global row tile base (rows 0..16383)

    v8f c[8];
    #pragma unroll
    for (int nt = 0; nt < 8; ++nt) c[nt] = (v8f){};

    const float* arow = hidden + (size_t)(m0 + ln) * H_;

    for (int k0 = 0; k0 < H_; k0 += 4) {
        // A 16x4 tile: VGPR0 = K=k0 (lanes0-15) / K=k0+2 (lanes16-31); VGPR1 = k0+1 / k0+3
        v2f a;
        a.x = arow[k0 + hi * 2];
        a.y = arow[k0 + hi * 2 + 1];
        // B 4x16 tiles: VGPR0 lanes0-15 = W[k0+0][n], lanes16-31 = W[k0+2][n]; VGPR1 = k0+1 / k0+3
        const float* wp = W + (size_t)(k0 + hi * 2) * N2D_ + ln;
        #pragma unroll
        for (int nt = 0; nt < 8; ++nt) {
            v2f bm;
            bm.x = wp[nt * 16];
            bm.y = wp[N2D_ + nt * 16];
            c[nt] = __builtin_amdgcn_wmma_f32_16x16x4_f32(
                false, a, false, bm, (short)0, c[nt], false, false);
        }
    }

    // Post: bias + RoPE + store. C layout: VGPR j: lanes0-15 = row M=j, col=ln;
    // lanes16-31 = row M=8+j, col=ln.
    const int bidx = m0 >> 9;            // batch
    const int s0   = m0 & (S_ - 1);      // sequence position base
    #pragma unroll
    for (int nt = 0; nt < 8; ++nt) {
        const int n   = nt * 16 + ln;    // output column 0..127
        const float bn = bias[n];
        const int d    = n & 63;         // column within q or k
        const int half = n >> 6;         // 0=q, 1=k
        const float inv_freq = expf(-(float)(d >> 1) * (9.210340371976184f / 32.0f));
        const bool even = ((n & 1) == 0);
        #pragma unroll
        for (int j = 0; j < 8; ++j) {
            float v = c[nt][j] + bn;
            float vpart = __shfl_xor(v, 1, 32);  // partner column (n^1), same row
            const int srow = s0 + j + hi * 8;    // sequence position
            const float ang = (float)srow * inv_freq;
            const float cc = cosf(ang), ss = sinf(ang);
            float r = even ? (v * cc - vpart * ss) : (v * cc + vpart * ss);
            const int grow = m0 + j + hi * 8;    // global row b*512+s
            if (half == 0)
                qw[(size_t)grow * D_ + d] = r;
            else
                kwT[((size_t)bidx * D_ + d) * S_ + srow] = r;
        }
    }
}

// ---------------------------------------------------------------------------
// Kernel 2: fused logits GEMM + masking + online double-logsumexp.
// One wave per (batch, 16-row tile). Never materializes logits.
// ---------------------------------------------------------------------------
__device__ __forceinline__ void lse_update(
    const v8f& c, int n0, int ln, int hi, int b, int s0,
    const float* __restrict__ mask, const int* __restrict__ span, size_t spanrow,
    float* mxn, float* smn, float* mxp, float* smp)
{
    const int n = n0 + ln;
    const float pad = mask[b * S_ + n];
    const float padoff = (1.0f - pad) * BIGC;
    #pragma unroll
    for (int j = 0; j < 8; ++j) {
        const int sm = s0 + j + hi * 8;
        float l = c[j] * pad - padoff;           // pad mask (ref order)
        if (sm > n) l -= BIGC;                   // strict lower-tri mask
        l *= 0.125f;                             // / sqrt(D)
        const int y = span[spanrow + (size_t)(j + hi * 8) * S_ + n];
        float yp = y ? -l : l;                   // (1-2y)*l exactly
        float vn = yp - (y ? BIGC : 0.0f);       // y_neg element
        float vp = yp - (y ? 0.0f : BIGC);       // y_pos element
        float nm = fmaxf(mxn[j], vn);
        smn[j] = smn[j] * expf(mxn[j] - nm) + expf(vn - nm);
        mxn[j] = nm;
        float pm = fmaxf(mxp[j], vp);
        smp[j] = smp[j] * expf(mxp[j] - pm) + expf(vp - pm);
        mxp[j] = pm;
    }
}

__global__ __launch_bounds__(32) void logits_loss_kernel(
    const float* __restrict__ qw,    // [B*S, 64]
    const float* __restrict__ kwT,   // [B, 64, 512]
    const float* __restrict__ mask,  // [B, S]
    const int* __restrict__ span,    // [B, S, S]
    float* __restrict__ rowloss)     // [B*S]
{
    const int lane = threadIdx.x;
    const int ln   = lane & 15;
    const int hi   = lane >> 4;
    const int m0   = blockIdx.x * 16;
    const int b    = m0 >> 9;
    const int s0   = m0 & (S_ - 1);
    const size_t spanrow = (size_t)m0 * S_;

    // Hold all A tiles (qw rows, K=64 => 16 k-steps of 4) in registers.
    v2f a[16];
    {
        const float* ap = qw + (size_t)(m0 + ln) * D_ + hi * 2;
        #pragma unroll
        for (int ks = 0; ks < 16; ++ks) { a[ks].x = ap[4 * ks]; a[ks].y = ap[4 * ks + 1]; }
    }

    // Running logsumexp state; init with the appended zero column: max=0, sum=1.
    float mxn[8], smn[8], mxp[8], smp[8];
    #pragma unroll
    for (int j = 0; j < 8; ++j) { mxn[j] = 0.f; smn[j] = 1.f; mxp[j] = 0.f; smp[j] = 1.f; }

    const float* kb = kwT + (size_t)b * D_ * S_;

    for (int ntb = 0; ntb < 32; ntb += 4) {  // 4 concurrent 16-col tiles for ILP
        const int n0 = ntb * 16;
        v8f c0 = (v8f){}, c1 = (v8f){}, c2 = (v8f){}, c3 = (v8f){};
        #pragma unroll
        for (int ks = 0; ks < 16; ++ks) {
            // B 4x16 tiles from kwT: lane-coalesced rows of 512.
            const float* bp = kb + (size_t)(4 * ks + hi * 2) * S_ + n0 + ln;
            v2f b0, b1, b2, b3;
            b0.x = bp[0];      b0.y = bp[S_];
            b1.x = bp[16];     b1.y = bp[S_ + 16];
            b2.x = bp[32];     b2.y = bp[S_ + 32];
            b3.x = bp[48];     b3.y = bp[S_ + 48];
            c0 = __builtin_amdgcn_wmma_f32_16x16x4_f32(false, a[ks], false, b0, (short)0, c0, false, false);
            c1 = __builtin_amdgcn_wmma_f32_16x16x4_f32(false, a[ks], false, b1, (short)0, c1, false, false);
            c2 = __builtin_amdgcn_wmma_f32_16x16x4_f32(false, a[ks], false, b2, (short)0, c2, false, false);
            c3 = __builtin_amdgcn_wmma_f32_16x16x4_f32(false, a[ks], false, b3, (short)0, c3, false, false);
        }
        lse_update(c0, n0,      ln, hi, b, s0, mask, span, spanrow, mxn, smn, mxp, smp);
        lse_update(c1, n0 + 16, ln, hi, b, s0, mask, span, spanrow, mxn, smn, mxp, smp);
        lse_update(c2, n0 + 32, ln, hi, b, s0, mask, span, spanrow, mxn, smn, mxp, smp);
        lse_update(c3, n0 + 48, ln, hi, b, s0, mask, span, spanrow, mxn, smn, mxp, smp);
    }

    // Merge across the 16 lanes of each half (butterfly stays within halves).
    #pragma unroll
    for (int j = 0; j < 8; ++j) {
        #pragma unroll
        for (int off = 1; off < 16; off <<= 1) {
            float om = __shfl_xor(mxn[j], off, 32);
            float os = __shfl_xor(smn[j], off, 32);
            float nm = fmaxf(mxn[j], om);
            smn[j] = smn[j] * expf(mxn[j] - nm) + os * expf(om - nm);
            mxn[j] = nm;
            om = __shfl_xor(mxp[j], off, 32);
            os = __shfl_xor(smp[j], off, 32);
            nm = fmaxf(mxp[j], om);
            smp[j] = smp[j] * expf(mxp[j] - nm) + os * expf(om - nm);
            mxp[j] = nm;
        }
        if (ln == 0) {
            float loss = mxn[j] + logf(smn[j]) + mxp[j] + logf(smp[j]);
            rowloss[m0 + j + hi * 8] = loss;
        }
    }
}

// ---------------------------------------------------------------------------
// Kernel 3: deterministic mean reduction of 16384 row losses -> scalar.
// ---------------------------------------------------------------------------
__global__ __launch_bounds__(256) void reduce_mean_kernel(
    const float* __restrict__ rowloss, float* __restrict__ out)
{
    __shared__ float sdata[256];
    float s = 0.0f;
    for (int i = threadIdx.x; i < B_ * S_; i += 256) s += rowloss[i];
    sdata[threadIdx.x] = s;
    __syncthreads();
    for (int off = 128; off > 0; off >>= 1) {
        if ((int)threadIdx.x < off) sdata[threadIdx.x] += sdata[threadIdx.x + off];
        __syncthreads();
    }
    if (threadIdx.x == 0) out[0] = sdata[0] * (1.0f / (float)(B_ * S_));
}

// ---------------------------------------------------------------------------
extern "C" void kernel_launch(void* const* d_in, const int* in_sizes, int n_in,
                              void* d_out, int out_size, void* d_ws, size_t ws_size,
                              hipStream_t stream) {
    (void)in_sizes; (void)n_in; (void)out_size; (void)ws_size;
    const float* hidden = (const float*)d_in[0];   // [B,S,H]
    const float* amask  = (const float*)d_in[1];   // [B,S]
    const int*   span   = (const int*)d_in[2];     // [B,S,S]
    const float* Wd     = (const float*)d_in[3];   // [H,128]
    const float* bd     = (const float*)d_in[4];   // [128]
    float* out = (float*)d_out;

    float* qw      = (float*)d_ws;                             // 16384*64
    float* kwT     = qw + (size_t)B_ * S_ * D_;                // 32*64*512
    float* rowloss = kwT + (size_t)B_ * D_ * S_;               // 16384

    proj_rope_kernel<<<(B_ * S_) / 16, 32, 0, stream>>>(hidden, Wd, bd, qw, kwT);
    logits_loss_kernel<<<(B_ * S_) / 16, 32, 0, stream>>>(qw, kwT, amask, span, rowloss);
    reduce_mean_kernel<<<1, 256, 0, stream>>>(rowloss, out);
}